// SoftRasterizer3D_5111011082242
// MI455X (gfx1250) — compile-verified
//
#include <hip/hip_runtime.h>

typedef __attribute__((ext_vector_type(2))) float v2f;
typedef __attribute__((ext_vector_type(8))) float v8f;

#define IMG 128
#define NVERT 256
#define NFACE 256
#define NBATCH 4
#define INV_SIGMA 10000.0f      /* 1/SIGMA, SIGMA=1e-4 */
#define INV_GAMMA 10.0f         /* 1/GAMMA, GAMMA=0.1  */
#define WBG 4.5399929762484854e-05f  /* exp(-1/GAMMA): fixed softmax shift m=1 */

__device__ __forceinline__ float clamp01(float x) { return fminf(fmaxf(x, 0.0f), 1.0f); }

// ---------------------------------------------------------------------------
// Workspace layout in d_ws:
//   geom[B][F][16] : 0:a0x 1:a0y 2:e01x 3:e01y 4:e12x 5:e12y 6:inv_denom
//                    7:vz0 8:vz1 9:vz2 10:ibb01 11:ibb12 12:ibb20 13..15:pad
//   brow[B][F][16] : [colR, colG, colB, 1, 0 x 12]  (pre-built WMMA B rows)
// ---------------------------------------------------------------------------
#define GEOM_FLOATS (NBATCH*NFACE*16)

__global__ void __launch_bounds__(256)
sr_project(const float* __restrict__ verts, const int* __restrict__ faces,
           const float* __restrict__ colors, const float* __restrict__ eye,
           float* __restrict__ fd)
{
    const int b = blockIdx.x;
    const int t = threadIdx.x;
    __shared__ float ndc[NVERT][3];

    // Camera basis (redundant per thread; trivial).  center=0, up=(0,1,0).
    float ex = eye[b*3+0], ey = eye[b*3+1], ez = eye[b*3+2];
    float zl = sqrtf(ex*ex + ey*ey + ez*ez);
    float zi = 1.0f / (zl + 1e-8f);
    float zx = ex*zi, zy = ey*zi, zz = ez*zi;
    float xl = sqrtf(zz*zz + zx*zx);
    float xi = 1.0f / (xl + 1e-8f);
    float xx = zz*xi, xy = 0.0f, xz = -zx*xi;
    float yx = zy*xz - zz*xy;
    float yy = zz*xx - zx*xz;
    float yz = zx*xy - zy*xx;
    float tx = -(xx*ex + xy*ey + xz*ez);
    float ty = -(yx*ex + yy*ey + yz*ez);
    float tz = -(zx*ex + zy*ey + zz*ez);
    const float fl  = 1.7320508075688772f;   // 1/tan(30deg)
    const float m22 = -1.0202020202020203f;  // (far+near)/(near-far)
    const float m23 = -0.20202020202020202f; // 2*far*near/(near-far)

    {
        float vx = verts[(b*NVERT + t)*3 + 0];
        float vy = verts[(b*NVERT + t)*3 + 1];
        float vz = verts[(b*NVERT + t)*3 + 2];
        float view_x = xx*vx + xy*vy + xz*vz + tx;
        float view_y = yx*vx + yy*vy + yz*vz + ty;
        float view_z = zx*vx + zy*vy + zz*vz + tz;
        float wi = 1.0f / (-view_z + 1e-8f);
        ndc[t][0] = fl*view_x*wi;
        ndc[t][1] = fl*view_y*wi;
        ndc[t][2] = (m22*view_z + m23)*wi;
    }
    __syncthreads();

    // Build face record t
    int i0 = faces[t*3+0], i1 = faces[t*3+1], i2 = faces[t*3+2];
    float a0x = ndc[i0][0], a0y = ndc[i0][1];
    float a1x = ndc[i1][0], a1y = ndc[i1][1];
    float a2x = ndc[i2][0], a2y = ndc[i2][1];
    float e01x = a1x - a0x, e01y = a1y - a0y;
    float e12x = a2x - a1x, e12y = a2y - a1y;
    float e20x = a0x - a2x, e20y = a0y - a2y;
    float area = e01x*(a2y - a0y) - e01y*(a2x - a0x);
    float denom = area + (area >= 0.0f ? 1e-12f : -1e-12f);

    float* g = fd + (b*NFACE + t)*16;
    g[0] = a0x;  g[1] = a0y;
    g[2] = e01x; g[3] = e01y;
    g[4] = e12x; g[5] = e12y;
    g[6] = 1.0f / denom;
    g[7] = ndc[i0][2]; g[8] = ndc[i1][2]; g[9] = ndc[i2][2];
    g[10] = 1.0f / (e01x*e01x + e01y*e01y + 1e-12f);
    g[11] = 1.0f / (e12x*e12x + e12y*e12y + 1e-12f);
    g[12] = 1.0f / (e20x*e20x + e20y*e20y + 1e-12f);
    g[13] = 0.0f; g[14] = 0.0f; g[15] = 0.0f;

    // Pre-built WMMA B row: [r, g, b, 1, 0 x 12]
    float* br = fd + GEOM_FLOATS + (b*NFACE + t)*16;
    br[0] = (colors[(b*NVERT+i0)*3+0] + colors[(b*NVERT+i1)*3+0] + colors[(b*NVERT+i2)*3+0]) * (1.0f/3.0f);
    br[1] = (colors[(b*NVERT+i0)*3+1] + colors[(b*NVERT+i1)*3+1] + colors[(b*NVERT+i2)*3+1]) * (1.0f/3.0f);
    br[2] = (colors[(b*NVERT+i0)*3+2] + colors[(b*NVERT+i1)*3+2] + colors[(b*NVERT+i2)*3+2]) * (1.0f/3.0f);
    br[3] = 1.0f;
    #pragma unroll
    for (int k = 4; k < 16; ++k) br[k] = 0.0f;
}

// ---------------------------------------------------------------------------
// Raster: one wave = one 16-pixel strip; faces accumulated 4 at a time through
// V_WMMA_F32_16X16X4_F32 (C cols: 0=R num, 1=G num, 2=B num, 3=softmax denom).
// Inner loop is fully uniform (EXEC all-1s around every WMMA).
// ---------------------------------------------------------------------------
__global__ void __launch_bounds__(256)
sr_raster(const float* __restrict__ fd, float* __restrict__ out)
{
    __shared__ float sg[NFACE * 16];           // 16 KB geometry records
    __shared__ float sb[NFACE * 16];           // 16 KB WMMA B rows
    const int b   = blockIdx.x >> 7;           // 128 blocks per batch
    const int blk = blockIdx.x & 127;
    const int tid = threadIdx.x;

    // Stage this batch's tables into LDS (1 face / thread, b128 traffic)
    {
        const float4* s0 = (const float4*)(fd + (b*NFACE + tid)*16);
        const float4* s1 = (const float4*)(fd + GEOM_FLOATS + (b*NFACE + tid)*16);
        float4* d0 = (float4*)(sg + tid*16);
        float4* d1 = (float4*)(sb + tid*16);
        d0[0] = s0[0]; d0[1] = s0[1]; d0[2] = s0[2]; d0[3] = s0[3];
        d1[0] = s1[0]; d1[1] = s1[1]; d1[2] = s1[2]; d1[3] = s1[3];
    }
    __syncthreads();

    const int wave = tid >> 5;
    const int lane = tid & 31;
    const int hi   = lane >> 4;                // half-wave: faces K+{0,1} vs K+{2,3}
    const int n    = lane & 15;                // A-matrix row (pixel) / B,C column
    const int strip = blk*8 + wave;            // 0..1023 per batch
    const int y    = strip >> 3;
    const int px0  = (strip & 7) << 4;

    const float pxc = ((float)(px0 + n) + 0.5f) * (1.0f/64.0f) - 1.0f;
    const float pyc = 1.0f - ((float)y + 0.5f) * (1.0f/64.0f);

    v8f c = {0.f,0.f,0.f,0.f,0.f,0.f,0.f,0.f};
    float omp = 1.0f;                          // running prod of (1 - prob)

    #pragma unroll 2
    for (int kb = 0; kb < NFACE; kb += 4) {
        v2f a, bm;
        #pragma unroll
        for (int j = 0; j < 2; ++j) {
            const int f = kb + hi*2 + j;       // matches A layout: K = j + 2*hi
            const float* F = sg + f*16;
            float a0x=F[0], a0y=F[1], e01x=F[2], e01y=F[3], e12x=F[4], e12y=F[5];
            float invd=F[6], vz0=F[7], vz1=F[8], vz2=F[9];
            float a1x = a0x + e01x, a1y = a0y + e01y;
            float a2x = a1x + e12x, a2y = a1y + e12y;
            float e20x = a0x - a2x, e20y = a0y - a2y;
            float p0x = pxc - a0x, p0y = pyc - a0y;
            float p1x = pxc - a1x, p1y = pyc - a1y;
            float p2x = pxc - a2x, p2y = pyc - a2y;
            // barycentrics
            float w0 = e12x*p1y - e12y*p1x;
            float w1 = e20x*p2y - e20y*p2x;
            float w2 = e01x*p0y - e01y*p0x;
            float b0 = w0*invd, b1 = w1*invd, b2 = w2*invd;
            bool inside = (b0 >= 0.0f) && (b1 >= 0.0f) && (b2 >= 0.0f);
            // min squared edge distance (sqrt cancels against d*d in sigmoid arg)
            float t0 = clamp01((p0x*e01x + p0y*e01y)*F[10]);
            float dx = p0x - t0*e01x, dy = p0y - t0*e01y;
            float dd = dx*dx + dy*dy;
            float t1 = clamp01((p1x*e12x + p1y*e12y)*F[11]);
            dx = p1x - t1*e12x; dy = p1y - t1*e12y;
            dd = fminf(dd, dx*dx + dy*dy);
            float t2 = clamp01((p2x*e20x + p2y*e20y)*F[12]);
            dx = p2x - t2*e20x; dy = p2y - t2*e20y;
            dd = fminf(dd, dx*dx + dy*dy);
            dd += 1e-12f;
            float xarg = (inside ? dd : -dd) * INV_SIGMA;
            // sigmoid via v_exp + v_rcp (saturates correctly: exp->inf => prob->0)
            float prob = __builtin_amdgcn_rcpf(1.0f + __expf(-xarg));
            omp *= (1.0f - prob);
            // soft-z (rcp instead of IEEE divide)
            float bc0 = clamp01(b0), bc1 = clamp01(b1), bc2 = clamp01(b2);
            float s = bc0 + bc1 + bc2 + 1e-8f;
            float zpix = (bc0*vz0 + bc1*vz1 + bc2*vz2) * __builtin_amdgcn_rcpf(s);
            float zn = clamp01((1.0f - zpix) * 0.5f);
            // softmax term with fixed shift m=1 (shift-invariant, zn<=1)
            a[j]  = prob * __expf((zn - 1.0f) * INV_GAMMA);   // A[M=n][K=2*hi+j]
            bm[j] = sb[f*16 + n];                             // B[K][N=n], branchless
        }
        // C += A(16x4 Dw) x B(4x16 [r g b 1 0..])   -> v_wmma_f32_16x16x4_f32
        c = __builtin_amdgcn_wmma_f32_16x16x4_f32(false, a, false, bm,
                                                  (short)0, c, false, false);
    }

    // ---- silhouette: combine half-wave partial products (faces 0,1 mod 4 vs 2,3)
    float ompAll = omp * __shfl_xor(omp, 16, 32);
    if (lane < 16)
        out[(b*IMG + y)*IMG + px0 + lane] = 1.0f - ompAll;

    // ---- rgb: C[j] holds (M = j + 8*hi, N = n); denominator lives at N==3
    float* rgb = out + NBATCH*IMG*IMG;
    #pragma unroll
    for (int j = 0; j < 8; ++j) {
        float sv  = __shfl(c[j], (lane & 16) | 3, 32);  // S for this row M
        float val = c[j] * __builtin_amdgcn_rcpf(sv + WBG);
        if (n < 3) {
            int m = j + 8*hi;
            rgb[((b*3 + n)*IMG + y)*IMG + px0 + m] = val;
        }
    }
}

extern "C" void kernel_launch(void* const* d_in, const int* in_sizes, int n_in,
                              void* d_out, int out_size, void* d_ws, size_t ws_size,
                              hipStream_t stream) {
    const float* verts  = (const float*)d_in[0];   // (4,256,3) f32
    const int*   faces  = (const int*)  d_in[1];   // (256,3)   i32
    const float* colors = (const float*)d_in[2];   // (4,256,3) f32
    const float* eye    = (const float*)d_in[3];   // (4,3)     f32
    float* out = (float*)d_out;                    // sil(4,1,128,128) ++ rgb(4,3,128,128)
    float* fd  = (float*)d_ws;                     // 2 * 64 KB

    sr_project<<<dim3(NBATCH), dim3(256), 0, stream>>>(verts, faces, colors, eye, fd);
    sr_raster <<<dim3(NBATCH*128), dim3(256), 0, stream>>>(fd, out);
}